// SO3LinearFlat_63986422776405
// MI455X (gfx1250) — compile-verified
//
#include <hip/hip_runtime.h>

typedef float v2f __attribute__((ext_vector_type(2)));
typedef float v8f __attribute__((ext_vector_type(8)));

#define IN_CH   128
#define OUT_CH  128
#define MDIM    25
#define BATCH   32768
#define ROW_STRIDE (MDIM * IN_CH)   // floats between consecutive b at fixed m
#define LDS_STRIDE 132              // 128 + 4 pad -> bank = 4*row + k (conflict-free)
#define TILES_PER_BLOCK 16          // 16 row-tiles x 16 rows = 256 rows per block

// gfx1250 async L2->LDS copy, tracked by ASYNCcnt (CDNA5 ISA)
__device__ __forceinline__ void async_copy_b128(uint32_t lds_byte, const float* gptr) {
    asm volatile("global_load_async_to_lds_b128 %0, %1, off"
                 :: "v"(lds_byte), "v"((unsigned long long)(uintptr_t)gptr)
                 : "memory");
}
__device__ __forceinline__ void wait_async_all() {
    asm volatile("s_wait_asynccnt 0" ::: "memory");
}

__global__ __launch_bounds__(256)
void so3_linear_wmma(const float* __restrict__ inp,
                     const float* __restrict__ weight,
                     const float* __restrict__ bias,
                     const int*   __restrict__ expand_index,
                     float* __restrict__ out)
{
    __shared__ float tile[2][16 * LDS_STRIDE];   // double-buffered staging (2 x 8448 B)

    const int m        = blockIdx.y;
    const int l        = expand_index[m];
    const int row_base = blockIdx.x * (16 * TILES_PER_BLOCK);

    const int lane  = threadIdx.x & 31;
    const int wid   = threadIdx.x >> 5;      // 0..7 -> which 16-wide N tile
    const int col   = lane & 15;             // N column / A row selector
    const int khi   = (lane >> 4) * 2;       // K sub-offset inside a K=4 step
    const int nbase = wid * 16;

    // ---- preload all B (weight) fragments for K=0..127 into registers ----
    // B[4x16] layout: lane L, vgpr j holds B[khi+j][L&15]
    const float* wl = weight + ((size_t)l * OUT_CH + (size_t)(nbase + col)) * IN_CH + khi;
    v2f wfrag[32];
#pragma unroll
    for (int k = 0; k < 32; ++k)
        wfrag[k] = *(const v2f*)(wl + k * 4);

    float bval = 0.0f;
    if (m == 0) bval = bias[nbase + col];    // block-uniform branch

    // staging coordinates: 256 lanes x 2 async b128 move one 16x128 f32 tile
    const int r  = threadIdx.x >> 5;         // 0..7
    const int c4 = (threadIdx.x & 31) * 4;   // float offset 0..124
    const uint32_t lds_base = (uint32_t)(size_t)(&tile[0][0]);
    const uint32_t lds_st0  = lds_base + (uint32_t)((r      * LDS_STRIDE + c4) * 4);
    const uint32_t lds_st1  = lds_base + (uint32_t)(((r + 8) * LDS_STRIDE + c4) * 4);
    const uint32_t buf_bytes = 16 * LDS_STRIDE * 4;

    // prologue: stage tile 0 into buffer 0
    {
        const float* src = inp + ((size_t)row_base * MDIM + m) * IN_CH;
        async_copy_b128(lds_st0, src + (size_t)r        * ROW_STRIDE + c4);
        async_copy_b128(lds_st1, src + (size_t)(r + 8)  * ROW_STRIDE + c4);
        wait_async_all();
        __syncthreads();
    }

    for (int mt = 0; mt < TILES_PER_BLOCK; ++mt) {
        const int cur      = mt & 1;
        const int tile_row = row_base + mt * 16;

        // issue async staging of the NEXT tile into the other buffer
        if (mt + 1 < TILES_PER_BLOCK) {
            const float* src = inp + ((size_t)(tile_row + 16) * MDIM + m) * IN_CH;
            const uint32_t nb = (1 - cur) * buf_bytes;
            async_copy_b128(lds_st0 + nb, src + (size_t)r       * ROW_STRIDE + c4);
            async_copy_b128(lds_st1 + nb, src + (size_t)(r + 8) * ROW_STRIDE + c4);
        }

        // ---- K sweep: 32 WMMAs, pipelined A-fragment loads, dual accumulators ----
        const float* lp = &tile[cur][col * LDS_STRIDE + khi];

        v2f a0 = *(const v2f*)(lp + 0);
        v2f a1 = *(const v2f*)(lp + 4);
        v2f a2 = *(const v2f*)(lp + 8);
        v2f a3 = *(const v2f*)(lp + 12);

        v8f acc0, acc1;
#pragma unroll
        for (int j = 0; j < 8; ++j) { acc0[j] = bval; acc1[j] = 0.0f; }

#pragma unroll
        for (int g = 0; g < 8; ++g) {
            v2f n0, n1, n2, n3;
            if (g < 7) {                      // prefetch next K=16 group
                const float* np = lp + (g + 1) * 16;
                n0 = *(const v2f*)(np + 0);
                n1 = *(const v2f*)(np + 4);
                n2 = *(const v2f*)(np + 8);
                n3 = *(const v2f*)(np + 12);
            }
            acc0 = __builtin_amdgcn_wmma_f32_16x16x4_f32(false, a0, false, wfrag[4*g+0],
                                                         (short)0, acc0, false, false);
            acc1 = __builtin_amdgcn_wmma_f32_16x16x4_f32(false, a1, false, wfrag[4*g+1],
                                                         (short)0, acc1, false, false);
            acc0 = __builtin_amdgcn_wmma_f32_16x16x4_f32(false, a2, false, wfrag[4*g+2],
                                                         (short)0, acc0, false, false);
            acc1 = __builtin_amdgcn_wmma_f32_16x16x4_f32(false, a3, false, wfrag[4*g+3],
                                                         (short)0, acc1, false, false);
            a0 = n0; a1 = n1; a2 = n2; a3 = n3;
        }
        const v8f acc = acc0 + acc1;

        // drain our async copies, then block-wide handoff to the next buffer
        if (mt + 1 < TILES_PER_BLOCK) {
            wait_async_all();
            __syncthreads();
        }

        // D layout: vgpr j, lane L -> row (j + (L>>4)*8), col nbase + (L&15)
        float* op = out + ((size_t)(tile_row + (lane >> 4) * 8) * MDIM + m) * OUT_CH
                        + nbase + col;
#pragma unroll
        for (int j = 0; j < 8; ++j)
            op[(size_t)j * MDIM * OUT_CH] = acc[j];
    }
}

extern "C" void kernel_launch(void* const* d_in, const int* in_sizes, int n_in,
                              void* d_out, int out_size, void* d_ws, size_t ws_size,
                              hipStream_t stream) {
    (void)in_sizes; (void)n_in; (void)d_ws; (void)ws_size; (void)out_size;
    const float* inp    = (const float*)d_in[0];
    const float* weight = (const float*)d_in[1];
    const float* bias   = (const float*)d_in[2];
    const int*   eidx   = (const int*)d_in[3];
    float*       out    = (float*)d_out;

    dim3 grid(BATCH / (16 * TILES_PER_BLOCK), MDIM, 1);  // (128, 25)
    so3_linear_wmma<<<grid, dim3(256, 1, 1), 0, stream>>>(inp, weight, bias, eidx, out);
}